// transforms_bayer_halftoning_71408126263541
// MI455X (gfx1250) — compile-verified
//
#include <hip/hip_runtime.h>

// Bayer ordered-dither halftoning, fused one-pass streaming kernel for MI455X.
// x: (32, 3, 1024, 1024) f32 -> out same. HBM-bound: 768 MiB traffic,
// floor ~34us @ 23.3 TB/s. Persistent grid-stride loop; per-iteration
// global_prefetch_b8 lookahead (one grid stride = 48 MiB working set << 192 MB L2);
// 128-bit non-temporal loads/stores; Bayer threshold computed arithmetically
// (closed form of the recursive matrix -> bitop3 fusion, no table gathers).

typedef float f4 __attribute__((ext_vector_type(4)));

__global__ __launch_bounds__(256) void bayer_halftone_kernel(
    const float* __restrict__ x, float* __restrict__ out, unsigned n_pix4)
{
    constexpr unsigned HW = 1u << 20;   // 1024*1024 plane
    constexpr unsigned W  = 1024u;

    const unsigned stride = gridDim.x * 256u;

    for (unsigned t = blockIdx.x * 256u + threadIdx.x; t < n_pix4; t += stride) {
        // --- prefetch next grid-stride iteration into L2 (gfx1250 global_prefetch_b8)
        unsigned tn = t + stride;
        if (tn < n_pix4) {
            unsigned pn    = tn << 2;
            unsigned basen = pn + ((pn >> 20) << 21);   // b*3*HW + r, c=0 plane
            __builtin_prefetch(x + basen,           0, 0);
            __builtin_prefetch(x + basen + HW,      0, 0);
            __builtin_prefetch(x + basen + 2u * HW, 0, 0);
        }

        // --- this iteration
        unsigned p = t << 2;                // flat pixel index over (b,h,w)
        unsigned b = p >> 20;               // image index
        unsigned r = p & (HW - 1u);         // index within plane
        unsigned h = r >> 10;
        unsigned w = r & (W - 1u);
        unsigned base = p + (b << 21);      // x[b,0,h,w] flat offset

        f4 a  = __builtin_nontemporal_load((const f4*)(x + base));            // c0, w=.114
        f4 bb = __builtin_nontemporal_load((const f4*)(x + base + HW));       // c1, w=.587
        f4 cc = __builtin_nontemporal_load((const f4*)(x + base + 2u * HW));  // c2, w=.299

        const unsigned j = h & 7u;   // bayer column index = h % 8
        f4 oa, ob, oc;
#pragma unroll
        for (int k = 0; k < 4; ++k) {
            // bayer row index i = w % 8; closed form of recursive Bayer matrix:
            // v = 16*f(i0,j0) + 4*f(i1,j1) + f(i2,j2), f(a,b) = ((a^b)<<1)|a
            unsigned i  = (w + (unsigned)k) & 7u;
            unsigned xr = i ^ j;
            unsigned v  = ((((xr      ) & 1u) << 1 | ((i      ) & 1u)) << 4)
                        + ((((xr >> 1u) & 1u) << 1 | ((i >> 1u) & 1u)) << 2)
                        + ((((xr >> 2u) & 1u) << 1 | ((i >> 2u) & 1u))     );
            float thresh = (float)v * (255.0f / 63.0f);

            float g = (a[k]  * 255.0f) * 0.114f
                    + (bb[k] * 255.0f) * 0.587f
                    + (cc[k] * 255.0f) * 0.299f;

            float ht = (g > thresh) ? 1.0f : 0.0f;
            oa[k] = (a[k]  == 0.0f) ? 0.0f : ht;
            ob[k] = (bb[k] == 0.0f) ? 0.0f : ht;
            oc[k] = (cc[k] == 0.0f) ? 0.0f : ht;
        }

        __builtin_nontemporal_store(oa, (f4*)(out + base));
        __builtin_nontemporal_store(ob, (f4*)(out + base + HW));
        __builtin_nontemporal_store(oc, (f4*)(out + base + 2u * HW));
    }
}

extern "C" void kernel_launch(void* const* d_in, const int* in_sizes, int n_in,
                              void* d_out, int out_size, void* d_ws, size_t ws_size,
                              hipStream_t stream)
{
    const float* x = (const float*)d_in[0];
    float* out = (float*)d_out;

    // 32*3*1024*1024 elements -> 33,554,432 pixels -> 8,388,608 float4-triples.
    unsigned n_pix4 = (unsigned)(in_sizes[0] / 12);

    // Persistent launch: 2048 blocks x 256 threads (8 wave32/block), 16 iters each.
    const unsigned n_blocks = 2048u;

    bayer_halftone_kernel<<<n_blocks, 256, 0, stream>>>(x, out, n_pix4);
}